// AttnDecoderRNN_4879082848490
// MI455X (gfx1250) — compile-verified
//
#include <hip/hip_runtime.h>
#include <math.h>

#define H 1024
#define V 50257
#define L 10

typedef float v2f __attribute__((ext_vector_type(2)));
typedef float v4f __attribute__((ext_vector_type(4)));
typedef float v8f __attribute__((ext_vector_type(8)));

__device__ __forceinline__ float waveReduceSum(float v) {
#pragma unroll
  for (int off = 16; off > 0; off >>= 1)
    v += __shfl_xor(v, off, 32);
  return v;
}

// ---------------------------------------------------------------------------
// Kernel 1: embedding + additive attention + attn_applied (single block).
// Writes concat2 = [embedded(H), attn_applied(H)] to ws[0..2H),
// attn_weights to out[V+H .. V+H+L).
// ---------------------------------------------------------------------------
__global__ __launch_bounds__(256) void k_attn(
    const int* __restrict__ input_idx, const float* __restrict__ hidden,
    const float* __restrict__ enc_outs, const float* __restrict__ emb,
    const float* __restrict__ attn_W, const float* __restrict__ attn_b,
    float* __restrict__ ws, float* __restrict__ out) {
  __shared__ float semb[H];
  __shared__ float red[256];
  __shared__ float saw[L];
  const int tid = threadIdx.x;
  const int row = input_idx[0];

  for (int i = tid; i < H; i += 256) {
    float e = emb[(size_t)row * H + i];
    semb[i] = e;
    ws[i] = e;  // concat2 first half = embedded
  }
  __syncthreads();

  // attention scores over concat1 = [embedded, h0]
  for (int l = 0; l < L; ++l) {
    float p = 0.f;
    for (int k = tid; k < 2 * H; k += 256) {
      float c = (k < H) ? semb[k] : hidden[k - H];
      p += c * attn_W[l * 2 * H + k];
    }
    red[tid] = p;
    __syncthreads();
    for (int s = 128; s > 0; s >>= 1) {
      if (tid < s) red[tid] += red[tid + s];
      __syncthreads();
    }
    if (tid == 0) saw[l] = red[0] + attn_b[l];
    __syncthreads();
  }

  // softmax over L (tiny, thread 0)
  if (tid == 0) {
    float m = saw[0];
    for (int l = 1; l < L; ++l) m = fmaxf(m, saw[l]);
    float s = 0.f;
    for (int l = 0; l < L; ++l) { saw[l] = expf(saw[l] - m); s += saw[l]; }
    float inv = 1.f / s;
    for (int l = 0; l < L; ++l) {
      saw[l] *= inv;
      out[V + H + l] = saw[l];  // attn_weights output
    }
  }
  __syncthreads();

  // attn_applied = attn_weights @ encoder_outputs
  for (int i = tid; i < H; i += 256) {
    float acc = 0.f;
#pragma unroll
    for (int l = 0; l < L; ++l) acc += saw[l] * enc_outs[l * H + i];
    ws[H + i] = acc;  // concat2 second half
  }
}

// ---------------------------------------------------------------------------
// Kernel 2: x = relu(comb_W @ concat2 + comb_b).  One wave per output row.
// ---------------------------------------------------------------------------
__global__ __launch_bounds__(256) void k_combine(
    const float* __restrict__ comb_W, const float* __restrict__ comb_b,
    const float* __restrict__ concat2, float* __restrict__ x_out) {
  const int wave = (blockIdx.x * 256 + threadIdx.x) >> 5;
  const int lane = threadIdx.x & 31;
  if (wave >= H) return;
  const float* wr = comb_W + (size_t)wave * (2 * H);
  float acc = 0.f;
  for (int k = lane * 4; k < 2 * H; k += 128) {
    v4f w = *(const v4f*)(wr + k);
    v4f c = *(const v4f*)(concat2 + k);
    acc += w.x * c.x + w.y * c.y + w.z * c.z + w.w * c.w;
  }
  acc = waveReduceSum(acc);
  if (lane == 0) x_out[wave] = fmaxf(acc + comb_b[wave], 0.f);
}

// ---------------------------------------------------------------------------
// Kernel 3: one GRU step (torch gate order r,z,n).  One wave per element j:
// six length-H dots, then gate math.  Writes h_new to out[V+j].
// ---------------------------------------------------------------------------
__global__ __launch_bounds__(256) void k_gru(
    const float* __restrict__ W_ih, const float* __restrict__ W_hh,
    const float* __restrict__ b_ih, const float* __restrict__ b_hh,
    const float* __restrict__ x, const float* __restrict__ hidden,
    float* __restrict__ out) {
  const int j = (blockIdx.x * 256 + threadIdx.x) >> 5;
  const int lane = threadIdx.x & 31;
  if (j >= H) return;
  const float* wi_r = W_ih + (size_t)j * H;
  const float* wi_z = W_ih + (size_t)(H + j) * H;
  const float* wi_n = W_ih + (size_t)(2 * H + j) * H;
  const float* wh_r = W_hh + (size_t)j * H;
  const float* wh_z = W_hh + (size_t)(H + j) * H;
  const float* wh_n = W_hh + (size_t)(2 * H + j) * H;

  float ar = 0.f, az = 0.f, an = 0.f, br = 0.f, bz = 0.f, bn = 0.f;
  for (int k = lane * 4; k < H; k += 128) {
    v4f xv = *(const v4f*)(x + k);
    v4f hv = *(const v4f*)(hidden + k);
    v4f w;
    w = *(const v4f*)(wi_r + k); ar += w.x*xv.x + w.y*xv.y + w.z*xv.z + w.w*xv.w;
    w = *(const v4f*)(wi_z + k); az += w.x*xv.x + w.y*xv.y + w.z*xv.z + w.w*xv.w;
    w = *(const v4f*)(wi_n + k); an += w.x*xv.x + w.y*xv.y + w.z*xv.z + w.w*xv.w;
    w = *(const v4f*)(wh_r + k); br += w.x*hv.x + w.y*hv.y + w.z*hv.z + w.w*hv.w;
    w = *(const v4f*)(wh_z + k); bz += w.x*hv.x + w.y*hv.y + w.z*hv.z + w.w*hv.w;
    w = *(const v4f*)(wh_n + k); bn += w.x*hv.x + w.y*hv.y + w.z*hv.z + w.w*hv.w;
  }
  ar = waveReduceSum(ar); az = waveReduceSum(az); an = waveReduceSum(an);
  br = waveReduceSum(br); bz = waveReduceSum(bz); bn = waveReduceSum(bn);
  if (lane == 0) {
    float r = 1.f / (1.f + expf(-((ar + b_ih[j]) + (br + b_hh[j]))));
    float z = 1.f / (1.f + expf(-((az + b_ih[H + j]) + (bz + b_hh[H + j]))));
    float n = tanhf((an + b_ih[2 * H + j]) + r * (bn + b_hh[2 * H + j]));
    out[V + j] = (1.f - z) * n + z * hidden[j];  // h_new output
  }
}

// ---------------------------------------------------------------------------
// Kernel 4: logits = out_W @ h_new + out_b via V_WMMA_F32_16X16X4_F32.
// Each wave owns 16 rows of out_W (A matrix).  B = h broadcast over all 16
// columns, so D[m, n] == y[m] for every n.  ISA A 16x4 layout: lanes 0-15
// hold (K0,K1), lanes 16-31 hold (K2,K3) -> a float2 load per lane, walking
// each row sequentially (cachelines fully consumed).  h staged in LDS.
// ---------------------------------------------------------------------------
__global__ __launch_bounds__(256) void k_proj(
    const float* __restrict__ out_W, const float* __restrict__ out_b,
    const float* __restrict__ h, float* __restrict__ out) {
  __shared__ float sh[H];
  const int tid = threadIdx.x;
  for (int i = tid; i < H; i += 256) sh[i] = h[i];
  __syncthreads();

  const int lane = tid & 31;
  const int tile = blockIdx.x * 128 + (tid >> 5) * 16;
  int row = tile + (lane & 15);
  if (row >= V) row = V - 1;  // keep EXEC all-1s through the WMMA loop
  const float* wr = out_W + (size_t)row * H;
  const int kk = (lane >> 4) << 1;  // 0 for lanes 0-15, 2 for lanes 16-31

  v8f c0 = {}; v8f c1 = {};
  for (int k = 0; k < H; k += 8) {
    v2f a0 = *(const v2f*)(wr + k + kk);
    v2f b0 = *(const v2f*)(sh + k + kk);
    c0 = __builtin_amdgcn_wmma_f32_16x16x4_f32(false, a0, false, b0,
                                               (short)0, c0, false, false);
    v2f a1 = *(const v2f*)(wr + k + 4 + kk);
    v2f b1 = *(const v2f*)(sh + k + 4 + kk);
    c1 = __builtin_amdgcn_wmma_f32_16x16x4_f32(false, a1, false, b1,
                                               (short)0, c1, false, false);
  }
  v8f c = c0 + c1;

  // D layout: VGPR r, lanes 0-15 => M=r; lanes 16-31 => M=8+r.  All N equal.
  if ((lane & 15) == 0) {
    const int mbase = tile + ((lane >> 4) << 3);
#pragma unroll
    for (int r = 0; r < 8; ++r) {
      int rr = mbase + r;
      if (rr < V) out[rr] = c[r] + out_b[rr];
    }
  }
}

// ---------------------------------------------------------------------------
// Kernel 5: lse = max + log(sum(exp(logit - max))) over V (single block).
// ---------------------------------------------------------------------------
__global__ __launch_bounds__(256) void k_lse(const float* __restrict__ logits,
                                             float* __restrict__ lse_out) {
  __shared__ float red[256];
  const int tid = threadIdx.x;
  float m = -INFINITY;
  for (int i = tid; i < V; i += 256) m = fmaxf(m, logits[i]);
  red[tid] = m;
  __syncthreads();
  for (int s = 128; s > 0; s >>= 1) {
    if (tid < s) red[tid] = fmaxf(red[tid], red[tid + s]);
    __syncthreads();
  }
  m = red[0];
  __syncthreads();
  float sum = 0.f;
  for (int i = tid; i < V; i += 256) sum += expf(logits[i] - m);
  red[tid] = sum;
  __syncthreads();
  for (int s = 128; s > 0; s >>= 1) {
    if (tid < s) red[tid] += red[tid + s];
    __syncthreads();
  }
  if (tid == 0) lse_out[0] = m + logf(red[0]);
}

// ---------------------------------------------------------------------------
// Kernel 6: log_softmax finalize: out[v] -= lse (in place; deterministic
// because k_proj rewrites every logit each call).
// ---------------------------------------------------------------------------
__global__ __launch_bounds__(256) void k_sub(float* __restrict__ out,
                                             const float* __restrict__ lse) {
  int i = blockIdx.x * 256 + threadIdx.x;
  if (i < V) out[i] -= lse[0];
}

// ---------------------------------------------------------------------------
extern "C" void kernel_launch(void* const* d_in, const int* in_sizes, int n_in,
                              void* d_out, int out_size, void* d_ws,
                              size_t ws_size, hipStream_t stream) {
  (void)in_sizes; (void)n_in; (void)out_size; (void)ws_size;
  const int*   input_idx = (const int*)d_in[0];
  const float* hidden    = (const float*)d_in[1];
  /* d_in[2] encoder_output is unused by the reference forward */
  const float* enc_outs  = (const float*)d_in[3];
  const float* emb       = (const float*)d_in[4];
  const float* attn_W    = (const float*)d_in[5];
  const float* attn_b    = (const float*)d_in[6];
  const float* comb_W    = (const float*)d_in[7];
  const float* comb_b    = (const float*)d_in[8];
  const float* W_ih      = (const float*)d_in[9];
  const float* W_hh      = (const float*)d_in[10];
  const float* b_ih      = (const float*)d_in[11];
  const float* b_hh      = (const float*)d_in[12];
  const float* out_W     = (const float*)d_in[13];
  const float* out_b     = (const float*)d_in[14];

  float* out = (float*)d_out;  // [logits(V) | h_new(H) | attn_weights(L)]
  float* ws  = (float*)d_ws;   // [concat2(2H) | x(H) | lse(1)]

  k_attn<<<1, 256, 0, stream>>>(input_idx, hidden, enc_outs, emb, attn_W,
                                attn_b, ws, out);
  k_combine<<<(H * 32) / 256, 256, 0, stream>>>(comb_W, comb_b, ws,
                                                ws + 2 * H);
  k_gru<<<(H * 32) / 256, 256, 0, stream>>>(W_ih, W_hh, b_ih, b_hh,
                                            ws + 2 * H, hidden, out);
  k_proj<<<(V + 127) / 128, 256, 0, stream>>>(out_W, out_b, out + V, out);
  k_lse<<<1, 256, 0, stream>>>(out, ws + 3 * H);
  k_sub<<<(V + 255) / 256, 256, 0, stream>>>(out, ws + 3 * H);
}